// kernel_generated_1_21887153341201
// MI455X (gfx1250) — compile-verified
//
#include <hip/hip_runtime.h>
#include <hip/hip_bf16.h>

// GEMM view of the op:  out[i, p] = sum_K Amat[i,K] * Bmat[K,p]
//   M = 128 (i), N = 784 (p = n*28+o), K = 192 (K = j*96 + l*3 + k)
//   Amat[i,K]   = w[i, l, j, k]
//   Bmat[K,p]   = (0 <= o+k-1 < 28) ? x[l, (2n+j-1) mod 56, (o+k-2) mod 28] : 0

typedef __attribute__((ext_vector_type(2))) float v2f;
typedef __attribute__((ext_vector_type(8))) float v8f;

#define M_DIM 128
#define N_DIM 784
#define K_DIM 192
#define A_ELEMS (M_DIM * K_DIM)   // 24576
#define B_ELEMS (K_DIM * N_DIM)   // 150528

// ---------------------------------------------------------------------------
// Stage 1: build im2col matrix B (192 x 784) and repacked weights A (128 x 192)
// ---------------------------------------------------------------------------
__global__ __launch_bounds__(256)
void im2col_pack(const float* __restrict__ x, const float* __restrict__ w,
                 float* __restrict__ Amat, float* __restrict__ Bmat) {
  int idx = blockIdx.x * blockDim.x + threadIdx.x;

  if (idx < A_ELEMS) {
    int i = idx / K_DIM, K = idx % K_DIM;
    int j = K / 96, rem = K % 96, l = rem / 3, k = rem % 3;
    // w flat: i*192 + l*6 + j*3 + k
    Amat[idx] = w[i * 192 + l * 6 + j * 3 + k];
  }

  int t = idx - A_ELEMS;
  if (t >= 0 && t < B_ELEMS) {
    int K = t / N_DIM, p = t % N_DIM;
    int j = K / 96, rem = K % 96, l = rem / 3, k = rem % 3;
    int n = p / 28, o = p % 28;
    int heff = o + k - 1;                    // unfold window position
    float val = 0.0f;
    if (heff >= 0 && heff < 28) {
      int s    = (2 * n + j + 55) % 56;      // roll(+1) on fused 56-dim
      int hcol = (heff + 27) % 28;           // roll(+1) on inner 28-dim
      val = x[l * 1568 + s * 28 + hcol];     // x[l, s, hcol]
    }
    Bmat[t] = val;
  }
}

// ---------------------------------------------------------------------------
// Stage 2: WMMA f32 GEMM. 1 wave32 = one 16x16 C tile; block = 4 waves sharing
// one 16-row A panel staged in LDS. K loop: 48 steps of V_WMMA_F32_16X16X4_F32.
// ---------------------------------------------------------------------------
__global__ __launch_bounds__(128)
void wmma_gemm(const float* __restrict__ Amat, const float* __restrict__ Bmat,
               float* __restrict__ out) {
  __shared__ float As[16 * K_DIM];           // 12 KB A panel for this M tile

  const int tid  = threadIdx.x;
  const int lane = tid & 31;
  const int wid  = tid >> 5;
  const int mt   = blockIdx.x / 13;          // 8 M-tiles
  const int ng   = blockIdx.x % 13;          // group of 4 N-tiles
  const int nt   = ng * 4 + wid;             // N-tile for this wave (valid < 49)
  const int m0   = mt * 16;

  // Cooperative, coalesced fill of the A panel (16 rows x 192 cols).
  for (int e = tid; e < 16 * K_DIM; e += 128) {
    As[e] = Amat[(m0 + (e / K_DIM)) * K_DIM + (e % K_DIM)];
  }
  __syncthreads();

  if (nt < 49) {
    const int m    = lane & 15;              // A-fragment row
    const int half = lane >> 4;              // K sub-slot (0: K+0/1, 1: K+2/3)
    const int n    = nt * 16 + (lane & 15);  // B/C column
    const float* Arow = &As[m * K_DIM + 2 * half];
    const float* Bcol = Bmat + n;

    v8f acc = {};                            // 16x16 f32 accumulator (8 VGPRs)

#pragma unroll
    for (int kk = 0; kk < K_DIM; kk += 4) {
      v2f av;                                // A[m][kk+2h], A[m][kk+2h+1]
      av.x = Arow[kk];
      av.y = Arow[kk + 1];
      v2f bv;                                // B[kk+2h][n], B[kk+2h+1][n]
      bv.x = Bcol[(kk + 2 * half) * N_DIM];
      bv.y = Bcol[(kk + 2 * half + 1) * N_DIM];
      // 8 args: (neg_a, A, neg_b, B, c_mod, C, reuse_a, reuse_b)
      acc = __builtin_amdgcn_wmma_f32_16x16x4_f32(
          false, av, false, bv, (short)0, acc, false, false);
    }

    // D layout: VGPR r of lane -> D[8*half + r][lane%16]
    const int mbase = m0 + 8 * half;
#pragma unroll
    for (int r = 0; r < 8; ++r) {
      out[(mbase + r) * N_DIM + n] = acc[r];
    }
  }
}

// ---------------------------------------------------------------------------
// Fallback (only if workspace is too small): direct scalar evaluation.
// ---------------------------------------------------------------------------
__global__ __launch_bounds__(256)
void naive_conv(const float* __restrict__ x, const float* __restrict__ w,
                float* __restrict__ out) {
  int idx = blockIdx.x * blockDim.x + threadIdx.x;   // over 128*784 outputs
  if (idx >= M_DIM * N_DIM) return;
  int i = idx / N_DIM, p = idx % N_DIM;
  int n = p / 28, o = p % 28;
  float acc = 0.0f;
  for (int j = 0; j < 2; ++j) {
    int s = (2 * n + j + 55) % 56;
    for (int l = 0; l < 32; ++l) {
      const float* xrow = x + l * 1568 + s * 28;
      const float* wrow = w + i * 192 + l * 6 + j * 3;
#pragma unroll
      for (int k = 0; k < 3; ++k) {
        int heff = o + k - 1;
        if (heff >= 0 && heff < 28) acc += xrow[(heff + 27) % 28] * wrow[k];
      }
    }
  }
  out[idx] = acc;
}

extern "C" void kernel_launch(void* const* d_in, const int* in_sizes, int n_in,
                              void* d_out, int out_size, void* d_ws, size_t ws_size,
                              hipStream_t stream) {
  const float* x = (const float*)d_in[0];   // (1,32,56,28)
  const float* w = (const float*)d_in[1];   // (128,32,2,3)
  float* out = (float*)d_out;               // (1,128,28,28)

  const size_t need = (size_t)(A_ELEMS + B_ELEMS) * sizeof(float);  // ~700 KB
  if (ws_size < need || d_ws == nullptr) {
    naive_conv<<<(M_DIM * N_DIM + 255) / 256, 256, 0, stream>>>(x, w, out);
    return;
  }

  float* Amat = (float*)d_ws;
  float* Bmat = (float*)((char*)d_ws + (size_t)A_ELEMS * sizeof(float));

  // Stage 1: (24576 + 150528) elements = 684 blocks of 256.
  im2col_pack<<<(A_ELEMS + B_ELEMS) / 256, 256, 0, stream>>>(x, w, Amat, Bmat);

  // Stage 2: 8 M-tiles x 13 N-tile groups = 104 blocks of 128 (4 waves).
  wmma_gemm<<<8 * 13, 128, 0, stream>>>(Amat, Bmat, out);
}